// GraphConv_7516192768197
// MI455X (gfx1250) — compile-verified
//
#include <hip/hip_runtime.h>

typedef float v2f __attribute__((ext_vector_type(2)));
typedef float v4f __attribute__((ext_vector_type(4)));
typedef float v8f __attribute__((ext_vector_type(8)));

#define N_NODES_C 100000
#define N_EDGES_C 1600000
#define D_C       128
#define LDS_PAD   132   // 128 + 4 floats: 16 rows hit distinct LDS banks
#define EDGES_PER_WAVE 512

// ---------------------------------------------------------------------------
// Kernel 1: w_input = x @ W^T using V_WMMA_F32_16X16X4_F32 (full fp32 path).
// Block = 256 threads = 8 waves. Block computes a 16-row x 128-col tile:
// wave j owns the 16-column block starting at 16*j, B fragments resident in
// 64 VGPRs, A tile staged once through LDS (bank-conflict-free pad).
// ---------------------------------------------------------------------------
__global__ __launch_bounds__(256) void gc_gemm_wmma(const float* __restrict__ x,
                                                    const float* __restrict__ w,
                                                    float* __restrict__ win) {
    __shared__ float ldsA[16 * LDS_PAD];
    const int tid  = threadIdx.x;
    const int lane = tid & 31;
    const int wave = tid >> 5;
    const int row0 = blockIdx.x * 16;

    // Cooperative load of the 16x128 A tile (2048 floats, 8 per thread).
    {
        const int f = tid * 8;            // stays within one row (8 | 128)
        const int r = f >> 7;
        const int c = f & 127;
        const v4f* src = (const v4f*)(x + (size_t)row0 * D_C + f);
        v4f a0 = src[0];
        v4f a1 = src[1];
        *(v4f*)(ldsA + r * LDS_PAD + c)     = a0;
        *(v4f*)(ldsA + r * LDS_PAD + c + 4) = a1;
    }

    // Preload B = W^T fragments for this wave's 16 columns, all K=128.
    // B (4x16, KxN): vgpr v holds K = 4*kk + v + 2*(lane>=16), N = lane&15.
    const int col0 = wave * 16;
    const int n    = lane & 15;
    const int khi  = (lane >> 4) * 2;
    v2f Bfrag[32];
#pragma unroll
    for (int kk = 0; kk < 32; ++kk) {
        Bfrag[kk] = *(const v2f*)(w + (size_t)(col0 + n) * D_C + kk * 4 + khi);
    }

    __syncthreads();

    v8f acc = {};
    const int m = lane & 15;
#pragma unroll
    for (int kk = 0; kk < 32; ++kk) {
        // A (16x4): vgpr v holds K = 4*kk + v + 2*(lane>=16), M = lane&15
        v2f a = *(const v2f*)(ldsA + m * LDS_PAD + kk * 4 + khi);
        acc = __builtin_amdgcn_wmma_f32_16x16x4_f32(
            /*neg_a=*/false, a, /*neg_b=*/false, Bfrag[kk],
            /*c_mod=*/(short)0, acc, /*reuse_a=*/false, /*reuse_b=*/false);
    }

    // C/D (16x16 f32): vgpr r holds M = r + 8*(lane>=16), N = lane&15.
    const int mBase = (lane >> 4) * 8;
#pragma unroll
    for (int r = 0; r < 8; ++r) {
        win[(size_t)(row0 + mBase + r) * D_C + col0 + n] = acc[r];
    }
}

// ---------------------------------------------------------------------------
// Kernel 0: zero the aggregation buffer (replay-safe re-init each call).
// ---------------------------------------------------------------------------
__global__ __launch_bounds__(256) void gc_zero(float* __restrict__ agg) {
    const int i4 = blockIdx.x * 256 + threadIdx.x;       // one float4 each
    if (i4 < (N_NODES_C * D_C) / 4) {
        v4f z = {};
        ((v4f*)agg)[i4] = z;
    }
}

// ---------------------------------------------------------------------------
// Kernel 2: SpMM. One wave per contiguous chunk of edges; whole wave works
// one edge (row/col/val are wave-uniform -> scalar loads), lane l handles
// output features [4l, 4l+4). adj_rows is sorted, so accumulate in registers
// and flush with f32 atomics only on row change (~16x atomic reduction).
// Gathers of w_input rows hit the 192MB L2 (w_input = 51MB).
// ---------------------------------------------------------------------------
__global__ __launch_bounds__(256) void gc_spmm(const int*   __restrict__ rows,
                                               const int*   __restrict__ cols,
                                               const float* __restrict__ vals,
                                               const float* __restrict__ win,
                                               float*       __restrict__ agg) {
    const int  lane = threadIdx.x & 31;
    const long wid  = (long)blockIdx.x * 8 + (threadIdx.x >> 5);
    long e0 = wid * EDGES_PER_WAVE;
    if (e0 >= N_EDGES_C) return;
    long e1 = e0 + EDGES_PER_WAVE;
    if (e1 > N_EDGES_C) e1 = N_EDGES_C;

    int cur = rows[e0];
    v4f acc = {};
    for (long e = e0; e < e1; ++e) {
        const int   r = rows[e];
        const int   c = cols[e];
        const float v = vals[e];
        if (r != cur) {
            float* dst = agg + (size_t)cur * D_C + lane * 4;
            __hip_atomic_fetch_add(dst + 0, acc.x, __ATOMIC_RELAXED, __HIP_MEMORY_SCOPE_AGENT);
            __hip_atomic_fetch_add(dst + 1, acc.y, __ATOMIC_RELAXED, __HIP_MEMORY_SCOPE_AGENT);
            __hip_atomic_fetch_add(dst + 2, acc.z, __ATOMIC_RELAXED, __HIP_MEMORY_SCOPE_AGENT);
            __hip_atomic_fetch_add(dst + 3, acc.w, __ATOMIC_RELAXED, __HIP_MEMORY_SCOPE_AGENT);
            acc = v4f{};
            cur = r;
        }
        // Prefetch a future gather row into cache (global_prefetch_b8).
        if (e + 8 < e1) {
            __builtin_prefetch(win + (size_t)cols[e + 8] * D_C + lane * 4, 0, 0);
        }
        const v4f w4 = *(const v4f*)(win + (size_t)c * D_C + lane * 4);
        acc += w4 * v;
    }
    float* dst = agg + (size_t)cur * D_C + lane * 4;
    __hip_atomic_fetch_add(dst + 0, acc.x, __ATOMIC_RELAXED, __HIP_MEMORY_SCOPE_AGENT);
    __hip_atomic_fetch_add(dst + 1, acc.y, __ATOMIC_RELAXED, __HIP_MEMORY_SCOPE_AGENT);
    __hip_atomic_fetch_add(dst + 2, acc.z, __ATOMIC_RELAXED, __HIP_MEMORY_SCOPE_AGENT);
    __hip_atomic_fetch_add(dst + 3, acc.w, __ATOMIC_RELAXED, __HIP_MEMORY_SCOPE_AGENT);
}

// ---------------------------------------------------------------------------
// Kernel 3: out = relu(agg + bias). Pure streaming, float4 vectorized.
// ---------------------------------------------------------------------------
__global__ __launch_bounds__(256) void gc_finalize(const float* __restrict__ agg,
                                                   const float* __restrict__ bias,
                                                   float* __restrict__ out) {
    const int i4 = blockIdx.x * 256 + threadIdx.x;
    if (i4 >= (N_NODES_C * D_C) / 4) return;
    v4f a = ((const v4f*)agg)[i4];
    v4f b = ((const v4f*)bias)[i4 & 31];     // (i4*4) % 128 == (i4 & 31)*4
    v4f s = a + b;
    s.x = fmaxf(s.x, 0.0f);
    s.y = fmaxf(s.y, 0.0f);
    s.z = fmaxf(s.z, 0.0f);
    s.w = fmaxf(s.w, 0.0f);
    ((v4f*)out)[i4] = s;
}

extern "C" void kernel_launch(void* const* d_in, const int* in_sizes, int n_in,
                              void* d_out, int out_size, void* d_ws, size_t ws_size,
                              hipStream_t stream) {
    const float* x     = (const float*)d_in[0];
    const int*   rows  = (const int*)  d_in[1];
    const int*   cols  = (const int*)  d_in[2];
    const float* vals  = (const float*)d_in[3];
    const float* w     = (const float*)d_in[4];
    const float* bias  = (const float*)d_in[5];

    float* win = (float*)d_out;   // w_input staged in d_out (dead after SpMM)
    float* agg = (float*)d_ws;    // 100000*128 floats = 51.2 MB in workspace

    const int elems4 = (N_NODES_C * D_C) / 4;            // 3,200,000
    const int zgrid  = (elems4 + 255) / 256;             // 12,500

    // 0) zero aggregation buffer
    gc_zero<<<zgrid, 256, 0, stream>>>(agg);

    // 1) w_input = x @ W^T  (WMMA fp32), 100000/16 = 6250 row tiles
    gc_gemm_wmma<<<N_NODES_C / 16, 256, 0, stream>>>(x, w, win);

    // 2) SpMM with sorted-row register accumulation + f32 atomics
    const long nWaves = (N_EDGES_C + EDGES_PER_WAVE - 1) / EDGES_PER_WAVE; // 3125
    const int  sgrid  = (int)((nWaves + 7) / 8);                           // 391
    gc_spmm<<<sgrid, 256, 0, stream>>>(rows, cols, vals, win, agg);

    // 3) out = relu(agg + bias)
    gc_finalize<<<zgrid, 256, 0, stream>>>(agg, bias, d_out ? (float*)d_out : nullptr);
}